// GNNModel_12773232738363
// MI455X (gfx1250) — compile-verified
//
#include <hip/hip_runtime.h>

typedef __attribute__((ext_vector_type(2))) float v2f;
typedef __attribute__((ext_vector_type(8))) float v8f;

// ------------------------------------------------------------------
// small utility kernels
// ------------------------------------------------------------------
__global__ void fill_kernel(float* __restrict__ p, float v, int n) {
    int i = blockIdx.x * blockDim.x + threadIdx.x;
    if (i < n) p[i] = v;
}

__global__ void degree_kernel(const int* __restrict__ dst, float* __restrict__ deg, int E) {
    int e = blockIdx.x * blockDim.x + threadIdx.x;
    if (e < E) atomicAdd(&deg[dst[e]], 1.0f);
}

__global__ void rsqrt_kernel(float* __restrict__ p, int n) {
    int i = blockIdx.x * blockDim.x + threadIdx.x;
    if (i < n) {
        float d = p[i];
        p[i] = (d > 0.0f) ? rsqrtf(d) : 0.0f;
    }
}

// ------------------------------------------------------------------
// WMMA GEMM:  Y[n x FOUT] = X[n x 128] * W[128 x FOUT]   (f32, full precision)
// 256 threads = 8 waves; each wave owns one 16-row tile and all FOUT/16
// column tiles. W staged in LDS in K-pair-interleaved layout so each B
// fragment is a single aligned ds_load_b64 straight into the even VGPR
// pair V_WMMA_F32_16X16X4_F32 needs (no register shuffles).
// Tail rows: clamped loads (EXEC must stay all-ones for WMMA), guarded stores.
// ------------------------------------------------------------------
template <int FOUT>
__global__ __launch_bounds__(256) void gemm_wmma_kernel(const float* __restrict__ X,
                                                        const float* __restrict__ W,
                                                        float* __restrict__ Y, int n) {
    // lwp[((k>>1)*FOUT + c)*2 + (k&1)] = W[k][c]
    __shared__ float lwp[128 * FOUT];
    for (int i = threadIdx.x; i < 128 * FOUT; i += 256) {
        int k = i / FOUT;
        int c = i - k * FOUT;
        lwp[(((k >> 1) * FOUT + c) << 1) + (k & 1)] = W[i];
    }
    __syncthreads();

    const int lane = threadIdx.x & 31;
    const int wv   = threadIdx.x >> 5;
    const int m0   = (blockIdx.x * 8 + wv) * 16;

    const int half = lane >> 4;   // 0: lanes 0-15 (K=0,1), 1: lanes 16-31 (K=2,3)
    const int lr   = lane & 15;
    const int koff = half * 2;

    int arow = m0 + lr;
    if (arow >= n) arow = n - 1;              // clamp, never predicate WMMA
    const float* __restrict__ xrow = X + (size_t)arow * 128 + koff;

    v8f acc[FOUT / 16];
#pragma unroll
    for (int j = 0; j < FOUT / 16; ++j) acc[j] = (v8f){};

    for (int k = 0; k < 128; k += 4) {
        v2f a = *(const v2f*)(xrow + k);      // A frag: K=k+koff, k+koff+1
        const float* __restrict__ brow = lwp + ((((k + koff) >> 1) * FOUT + lr) << 1);
#pragma unroll
        for (int j = 0; j < FOUT / 16; ++j) {
            v2f b = *(const v2f*)(brow + (j << 5));   // {W[k+koff][c], W[k+koff+1][c]}
            acc[j] = __builtin_amdgcn_wmma_f32_16x16x4_f32(
                false, a, false, b, (short)0, acc[j], false, false);
        }
    }

    const int rbase = m0 + half * 8;          // C/D: VGPR r -> rows r / r+8
#pragma unroll
    for (int j = 0; j < FOUT / 16; ++j) {
#pragma unroll
        for (int r = 0; r < 8; ++r) {
            int row = rbase + r;
            if (row < n) Y[(size_t)row * FOUT + j * 16 + lr] = acc[j][r];
        }
    }
}

// ------------------------------------------------------------------
// self-loop + bias:  out[i,c] = xw[i,c] * dinv[i]^2 + b[c]   (single writer)
// ------------------------------------------------------------------
template <int F>
__global__ void init_agg_kernel(const float* __restrict__ xw, const float* __restrict__ dinv,
                                const float* __restrict__ bias, float* __restrict__ out, int n) {
    size_t i = (size_t)blockIdx.x * blockDim.x + threadIdx.x;
    if (i >= (size_t)n * F) return;
    int node = (int)(i / F);
    int c    = (int)(i % F);
    float di = dinv[node];
    out[i] = xw[i] * di * di + bias[c];
}

// ------------------------------------------------------------------
// edge scatter: one wave per edge, lanes stride channels (coalesced
// gather + coalesced f32 atomic adds)
// ------------------------------------------------------------------
template <int F>
__global__ void edge_agg_kernel(const int* __restrict__ src, const int* __restrict__ dst,
                                const float* __restrict__ dinv, const float* __restrict__ xw,
                                float* __restrict__ out, int E) {
    int wid  = (int)(((size_t)blockIdx.x * blockDim.x + threadIdx.x) >> 5);
    int lane = threadIdx.x & 31;
    if (wid >= E) return;
    int s = src[wid];
    int d = dst[wid];
    float w = dinv[s] * dinv[d];
#pragma unroll
    for (int c = 0; c < F; c += 32) {
        float v = xw[(size_t)s * F + c + lane] * w;
        atomicAdd(out + (size_t)d * F + c + lane, v);
    }
}

// ------------------------------------------------------------------
// BatchNorm: per-channel sum / sumsq (128 channels fixed)
// ------------------------------------------------------------------
__global__ void bn_stats_kernel(const float* __restrict__ h, float* __restrict__ stats, int n) {
    int c = threadIdx.x;                       // blockDim.x == 128
    float s = 0.0f, q = 0.0f;
    for (int r = blockIdx.x; r < n; r += gridDim.x) {
        float v = h[(size_t)r * 128 + c];
        s += v;
        q += v * v;
    }
    atomicAdd(&stats[c], s);
    atomicAdd(&stats[128 + c], q);
}

__global__ void bn_finalize_kernel(const float* __restrict__ stats, const float* __restrict__ g,
                                   const float* __restrict__ be, float* __restrict__ ss,
                                   float inv_n) {
    int c = threadIdx.x;                       // blockDim.x == 128
    float mu  = stats[c] * inv_n;
    float var = stats[128 + c] * inv_n - mu * mu;
    float sc  = g[c] * rsqrtf(var + 1e-5f);
    ss[c]       = sc;
    ss[128 + c] = be[c] - mu * sc;
}

__global__ void bn_relu_kernel(float* __restrict__ h, const float* __restrict__ ss, int n) {
    size_t i = (size_t)blockIdx.x * blockDim.x + threadIdx.x;
    if (i >= (size_t)n * 128) return;
    int c = (int)(i & 127);
    float v = h[i] * ss[c] + ss[128 + c];
    h[i] = v > 0.0f ? v : 0.0f;
}

// ------------------------------------------------------------------
// launch
// ------------------------------------------------------------------
extern "C" void kernel_launch(void* const* d_in, const int* in_sizes, int n_in,
                              void* d_out, int out_size, void* d_ws, size_t ws_size,
                              hipStream_t stream) {
    const float* x   = (const float*)d_in[0];
    const int*   ei  = (const int*)d_in[1];     // [2, E] : src row then dst row
    const float* W1  = (const float*)d_in[2];
    const float* b1  = (const float*)d_in[3];
    const float* g1  = (const float*)d_in[4];
    const float* be1 = (const float*)d_in[5];
    const float* W2  = (const float*)d_in[6];
    const float* b2  = (const float*)d_in[7];
    const float* g2  = (const float*)d_in[8];
    const float* be2 = (const float*)d_in[9];
    const float* W3  = (const float*)d_in[10];
    const float* b3  = (const float*)d_in[11];
    float* out = (float*)d_out;

    const int N = in_sizes[0] / 128;
    const int E = in_sizes[1] / 2;
    const int* srcIdx = ei;
    const int* dstIdx = ei + E;

    float* ws    = (float*)d_ws;
    float* dinv  = ws;                          // N
    float* bufA  = dinv + N;                    // N*128 (xw)
    float* bufB  = bufA + (size_t)N * 128;      // N*128 (agg / activations)
    float* stats = bufB + (size_t)N * 128;      // 256 (sum, sumsq)
    float* ss    = stats + 256;                 // 256 (scale, shift)

    const int TB = 256;
    const int NF = N * 128;
    const int NO = N * 64;
    const int gemmBlocks = (N + 127) / 128;     // 8 waves x 16 rows per block
    const int edgeBlocks = (E + 7) / 8;         // one wave per edge
    const float invN = 1.0f / (float)N;

    // --- symmetric GCN normalization: dinv = rsqrt(indeg + 1) ---
    fill_kernel<<<(N + TB - 1) / TB, TB, 0, stream>>>(dinv, 1.0f, N);
    degree_kernel<<<(E + TB - 1) / TB, TB, 0, stream>>>(dstIdx, dinv, E);
    rsqrt_kernel<<<(N + TB - 1) / TB, TB, 0, stream>>>(dinv, N);

    // --- layer 1 ---
    gemm_wmma_kernel<128><<<gemmBlocks, 256, 0, stream>>>(x, W1, bufA, N);
    init_agg_kernel<128><<<(NF + TB - 1) / TB, TB, 0, stream>>>(bufA, dinv, b1, bufB, N);
    edge_agg_kernel<128><<<edgeBlocks, 256, 0, stream>>>(srcIdx, dstIdx, dinv, bufA, bufB, E);
    fill_kernel<<<1, 256, 0, stream>>>(stats, 0.0f, 256);
    bn_stats_kernel<<<1024, 128, 0, stream>>>(bufB, stats, N);
    bn_finalize_kernel<<<1, 128, 0, stream>>>(stats, g1, be1, ss, invN);
    bn_relu_kernel<<<(NF + TB - 1) / TB, TB, 0, stream>>>(bufB, ss, N);

    // --- layer 2 ---
    gemm_wmma_kernel<128><<<gemmBlocks, 256, 0, stream>>>(bufB, W2, bufA, N);
    init_agg_kernel<128><<<(NF + TB - 1) / TB, TB, 0, stream>>>(bufA, dinv, b2, bufB, N);
    edge_agg_kernel<128><<<edgeBlocks, 256, 0, stream>>>(srcIdx, dstIdx, dinv, bufA, bufB, E);
    fill_kernel<<<1, 256, 0, stream>>>(stats, 0.0f, 256);
    bn_stats_kernel<<<1024, 128, 0, stream>>>(bufB, stats, N);
    bn_finalize_kernel<<<1, 128, 0, stream>>>(stats, g2, be2, ss, invN);
    bn_relu_kernel<<<(NF + TB - 1) / TB, TB, 0, stream>>>(bufB, ss, N);

    // --- layer 3 (no BN/ReLU, output N x 64) ---
    gemm_wmma_kernel<64><<<gemmBlocks, 256, 0, stream>>>(bufB, W3, bufA, N);
    init_agg_kernel<64><<<(NO + TB - 1) / TB, TB, 0, stream>>>(bufA, dinv, b3, out, N);
    edge_agg_kernel<64><<<edgeBlocks, 256, 0, stream>>>(srcIdx, dstIdx, dinv, bufA, out, E);
}